// FeatureVoxelConcatenation_20521353740730
// MI455X (gfx1250) — compile-verified
//
#include <hip/hip_runtime.h>

#define BATCH 8
#define CH 64
#define NPTS 65536
#define RRES 32
#define R3 (RRES*RRES*RRES)
#define NCHUNK 16
#define CHUNK (NPTS/NCHUNK)   // 4096

typedef __attribute__((ext_vector_type(4))) unsigned int u32x4;
typedef __attribute__((ext_vector_type(8))) unsigned int u32x8;

// ---------------------------------------------------------------------------
// Tensor Data Mover: async 2D tile load Global -> LDS (CDNA5, ISA ch.8).
// 2-group descriptor (2D tensor, VADDR2/3 = NULL). Issued once per calling
// wave (EXEC ignored) -- caller must guard to a single wave per block.
// LDS padding: +1 dword after every 64 dwords -> rows of 65 dwords, so a
// channel-major LDS tile can be read point-major without bank conflicts.
// ---------------------------------------------------------------------------
__device__ __forceinline__ void tdm_load_2d_tile_f32(
        unsigned lds_byte_off, const float* gptr,
        unsigned tensor_d0, unsigned tensor_d1,
        unsigned tile_d0, unsigned tile_d1, unsigned row_stride_elems) {
    const unsigned long long ga = (unsigned long long)gptr;
    u32x4 g0;
    g0[0] = 1u;                                  // count=1 (valid), load, no gather
    g0[1] = lds_byte_off;                        // lds_addr (bytes)
    g0[2] = (unsigned)ga;                        // global_addr[31:0]
    g0[3] = (unsigned)((ga >> 32) & 0x1FFFFFFu)  // global_addr[56:32]
          | (2u << 30);                          // type = 2 ("image")
    u32x8 g1;
    g1[0] = (2u << 16)                           // data_size = 4 bytes
          | (1u << 20)                           // pad_enable
          | (5u << 22);                          // pad_interval: every 64 dwords
                                                 // pad_amount code 0 = 1 dword
    g1[1] = (tensor_d0 & 0xFFFFu) << 16;                               // dim0 lo16
    g1[2] = (tensor_d0 >> 16) | ((tensor_d1 & 0xFFFFu) << 16);         // dim0 hi / dim1 lo
    g1[3] = (tensor_d1 >> 16) | (tile_d0 << 16);                       // dim1 hi / tile_dim0
    g1[4] = tile_d1;                                                   // tile_dim1 (tile_dim2=0)
    g1[5] = row_stride_elems;                                          // dim0_stride[31:0]
    g1[6] = 0u;                                                        // dim0_stride hi / dim1_stride lo
    g1[7] = 0u;
    asm volatile("tensor_load_to_lds %0, %1" :: "s"(g0), "s"(g1) : "memory");
}

// ---------------------------------------------------------------------------
// Stage 1a: partial sums for the coordinate means (deterministic tree reduce)
// ---------------------------------------------------------------------------
__global__ void k_partial_sum(const float* __restrict__ c1,
                              const float* __restrict__ c2,
                              float* __restrict__ partial) {
    const int inb   = blockIdx.x >> 4;   // 0..15 : input*8 + b
    const int chunk = blockIdx.x & 15;
    const int input = inb >> 3;
    const int b     = inb & 7;
    const float* base = ((input == 0) ? c1 : c2) + (size_t)b * 3 * NPTS + chunk * CHUNK;

    float sx = 0.f, sy = 0.f, sz = 0.f;
    for (int n = threadIdx.x; n < CHUNK; n += 256) {
        sx += base[n];
        sy += base[NPTS + n];
        sz += base[2 * NPTS + n];
    }
    __shared__ float red[3][256];
    red[0][threadIdx.x] = sx; red[1][threadIdx.x] = sy; red[2][threadIdx.x] = sz;
    __syncthreads();
    for (int s = 128; s > 0; s >>= 1) {
        if (threadIdx.x < s) {
            red[0][threadIdx.x] += red[0][threadIdx.x + s];
            red[1][threadIdx.x] += red[1][threadIdx.x + s];
            red[2][threadIdx.x] += red[2][threadIdx.x + s];
        }
        __syncthreads();
    }
    if (threadIdx.x == 0) {
        partial[(inb * NCHUNK + chunk) * 4 + 0] = red[0][0];
        partial[(inb * NCHUNK + chunk) * 4 + 1] = red[1][0];
        partial[(inb * NCHUNK + chunk) * 4 + 2] = red[2][0];
    }
}

__global__ void k_mean_final(const float* __restrict__ partial,
                             float* __restrict__ stats) {
    const int t = threadIdx.x;
    if (t < 16) {
        float sx = 0.f, sy = 0.f, sz = 0.f;
        for (int k = 0; k < NCHUNK; ++k) {
            sx += partial[(t * NCHUNK + k) * 4 + 0];
            sy += partial[(t * NCHUNK + k) * 4 + 1];
            sz += partial[(t * NCHUNK + k) * 4 + 2];
        }
        const float inv = 1.0f / (float)NPTS;
        stats[t * 4 + 0] = sx * inv;
        stats[t * 4 + 1] = sy * inv;
        stats[t * 4 + 2] = sz * inv;
    }
}

// ---------------------------------------------------------------------------
// Stage 1b: scale = 2 * max ||c - mean||
// ---------------------------------------------------------------------------
__global__ void k_partial_max(const float* __restrict__ c1,
                              const float* __restrict__ c2,
                              const float* __restrict__ stats,
                              float* __restrict__ partial) {
    const int inb   = blockIdx.x >> 4;
    const int chunk = blockIdx.x & 15;
    const int input = inb >> 3;
    const int b     = inb & 7;
    const float* base = ((input == 0) ? c1 : c2) + (size_t)b * 3 * NPTS + chunk * CHUNK;
    const float mx = stats[inb * 4 + 0];
    const float my = stats[inb * 4 + 1];
    const float mz = stats[inb * 4 + 2];

    float mm = 0.f;
    for (int n = threadIdx.x; n < CHUNK; n += 256) {
        float dx = base[n] - mx;
        float dy = base[NPTS + n] - my;
        float dz = base[2 * NPTS + n] - mz;
        mm = fmaxf(mm, sqrtf(dx * dx + dy * dy + dz * dz));
    }
    __shared__ float red[256];
    red[threadIdx.x] = mm;
    __syncthreads();
    for (int s = 128; s > 0; s >>= 1) {
        if (threadIdx.x < s) red[threadIdx.x] = fmaxf(red[threadIdx.x], red[threadIdx.x + s]);
        __syncthreads();
    }
    if (threadIdx.x == 0) partial[(inb * NCHUNK + chunk) * 4 + 0] = red[0];
}

__global__ void k_max_final(const float* __restrict__ partial,
                            float* __restrict__ stats) {
    const int t = threadIdx.x;
    if (t < 16) {
        float m = 0.f;
        for (int k = 0; k < NCHUNK; ++k)
            m = fmaxf(m, partial[(t * NCHUNK + k) * 4 + 0]);
        stats[t * 4 + 3] = 2.0f * m;
    }
}

// ---------------------------------------------------------------------------
// Stage 2: zero voxel sums + counts (ws is poisoned, must re-init every call)
// ---------------------------------------------------------------------------
__global__ void k_zero(float4* __restrict__ vox4, float* __restrict__ cnt) {
    const size_t i = (size_t)blockIdx.x * blockDim.x + threadIdx.x;
    const size_t NV4 = (size_t)BATCH * R3 * CH / 4;
    if (i < NV4) vox4[i] = make_float4(0.f, 0.f, 0.f, 0.f);
    if (i < (size_t)BATCH * R3) cnt[i] = 0.f;
}

// ---------------------------------------------------------------------------
// Stage 3: avg_voxelize(x2) scatter. vox layout: [b][voxel][c] (channel fast)
// Block: 256 thr, tile = 64 points of one batch. The 64ch x 64pt feature tile
// is DMA'd into LDS by the TDM (one tensor_load_to_lds per block, wave 0),
// overlapped with voxel-index computation; atomics go out as 256B-contiguous
// bursts per point into the L2-resident grid.
// ---------------------------------------------------------------------------
__global__ void k_scatter(const float* __restrict__ feat2,
                          const float* __restrict__ coords2,
                          const float* __restrict__ stats,
                          float* __restrict__ vox,
                          float* __restrict__ cnt) {
    const int b   = blockIdx.y;
    const int n0  = blockIdx.x * 64;
    const int tid = threadIdx.x;

    __shared__ float ldsT[CH][CH + 1];   // [channel][point] + TDM pad dword
    __shared__ int   vidx[64];

    // wave 0: kick off the async tile DMA (global row stride = NPTS floats)
    if (tid < 32) {
        tdm_load_2d_tile_f32((unsigned)(unsigned long long)&ldsT[0][0],
                             feat2 + ((size_t)b * CH) * NPTS + n0,
                             /*tensor_d0=*/NPTS, /*tensor_d1=*/CH,
                             /*tile_d0=*/64, /*tile_d1=*/CH,
                             /*row_stride=*/NPTS);
    }

    const float4 st = ((const float4*)stats)[8 + b];   // x2 stats
    const float invscale = 1.0f / st.w;

    if (tid < 64) {
        const int n = n0 + tid;
        float cx = coords2[((size_t)b * 3 + 0) * NPTS + n];
        float cy = coords2[((size_t)b * 3 + 1) * NPTS + n];
        float cz = coords2[((size_t)b * 3 + 2) * NPTS + n];
        float nx = fminf(fmaxf(((cx - st.x) * invscale + 0.5f) * (float)RRES, 0.f), 31.f);
        float ny = fminf(fmaxf(((cy - st.y) * invscale + 0.5f) * (float)RRES, 0.f), 31.f);
        float nz = fminf(fmaxf(((cz - st.z) * invscale + 0.5f) * (float)RRES, 0.f), 31.f);
        int ix = (int)rintf(nx), iy = (int)rintf(ny), iz = (int)rintf(nz);  // RNE = jnp.round
        vidx[tid] = (ix * RRES + iy) * RRES + iz;
    }

    if (tid < 32) __builtin_amdgcn_s_wait_tensorcnt(0);  // only wave 0 has TENSORcnt
    __syncthreads();

    const int g = tid >> 6;   // 4 point-subgroups of 64 channel-threads
    const int c = tid & 63;
    for (int p = g; p < 64; p += 4) {
        float* dst = vox + ((size_t)b * R3 + vidx[p]) * CH;
        atomicAdd(dst + c, ldsT[c][p]);               // 256B contiguous burst / point
        if (c == 0) atomicAdd(cnt + (size_t)b * R3 + vidx[p], 1.0f);
    }
}

// ---------------------------------------------------------------------------
// Stage 4: trilinear devoxelize at x1 coords (1/cnt folded into weights),
// transposed store through LDS, fused concat-copy of x1 features (prefetched).
// ---------------------------------------------------------------------------
__global__ void k_gather(const float* __restrict__ feat1,
                         const float* __restrict__ coords1,
                         const float* __restrict__ stats,
                         const float* __restrict__ vox,
                         const float* __restrict__ cnt,
                         float* __restrict__ out) {
    const int b   = blockIdx.y;
    const int n0  = blockIdx.x * 64;
    const int tid = threadIdx.x;

    __shared__ float tile[64][CH + 1];
    __shared__ int   cidx[64][8];
    __shared__ float cwt[64][8];

    // prefetch the x1-feature tile for the concat copy (global_prefetch_b8)
    {
        const float* src = feat1 + (size_t)b * CH * NPTS + n0;
        if (tid < 128) {
            const int ch = tid >> 1, half = tid & 1;
            __builtin_prefetch(src + (size_t)ch * NPTS + half * 32, 0, 3);
        }
    }

    const float4 st = ((const float4*)stats)[b];       // x1 stats
    const float invscale = 1.0f / st.w;
    const size_t basev = (size_t)b * R3;

    if (tid < 64) {
        const int n = n0 + tid;
        float cx = coords1[((size_t)b * 3 + 0) * NPTS + n];
        float cy = coords1[((size_t)b * 3 + 1) * NPTS + n];
        float cz = coords1[((size_t)b * 3 + 2) * NPTS + n];
        float nx = fminf(fmaxf(((cx - st.x) * invscale + 0.5f) * (float)RRES, 0.f), 31.f);
        float ny = fminf(fmaxf(((cy - st.y) * invscale + 0.5f) * (float)RRES, 0.f), 31.f);
        float nz = fminf(fmaxf(((cz - st.z) * invscale + 0.5f) * (float)RRES, 0.f), 31.f);
        float lx = floorf(nx), ly = floorf(ny), lz = floorf(nz);
        float dx = nx - lx,   dy = ny - ly,   dz = nz - lz;
        int ix0 = (int)lx, iy0 = (int)ly, iz0 = (int)lz;
        int ix1 = min(ix0 + 1, RRES - 1);
        int iy1 = min(iy0 + 1, RRES - 1);
        int iz1 = min(iz0 + 1, RRES - 1);
#pragma unroll
        for (int k = 0; k < 8; ++k) {
            const int xi = (k & 4) ? ix1 : ix0;
            const int yi = (k & 2) ? iy1 : iy0;
            const int zi = (k & 1) ? iz1 : iz0;
            const float wx = (k & 4) ? dx : 1.f - dx;
            const float wy = (k & 2) ? dy : 1.f - dy;
            const float wz = (k & 1) ? dz : 1.f - dz;
            const int idx = (xi * RRES + yi) * RRES + zi;
            const float cc = cnt[basev + idx];
            cidx[tid][k] = idx;
            cwt[tid][k]  = wx * wy * wz / fmaxf(cc, 1.0f);
        }
    }
    __syncthreads();

    const int g = tid >> 6;
    const int c = tid & 63;
    for (int p = g; p < 64; p += 4) {
        float acc = 0.f;
#pragma unroll
        for (int k = 0; k < 8; ++k) {
            // 64 threads read 256B contiguous (2x128B wave32 loads), L2-resident
            acc += cwt[p][k] * vox[(basev + (size_t)cidx[p][k]) * CH + c];
        }
        tile[p][c] = acc;
    }
    __syncthreads();

    // devox half of the output: coalesced channel-major stores
    for (int e = tid; e < 64 * CH; e += 256) {
        const int p  = e & 63;
        const int ch = e >> 6;
        out[((size_t)b * 2 * CH + CH + ch) * NPTS + n0 + p] = tile[p][ch];
    }
    // concat copy of x1 features (float4, fully coalesced)
    for (int e = tid; e < 64 * CH / 4; e += 256) {
        const int ch = e >> 4;
        const int pv = e & 15;
        const float4 v = ((const float4*)(feat1 + ((size_t)b * CH + ch) * NPTS + n0))[pv];
        ((float4*)(out + ((size_t)b * 2 * CH + ch) * NPTS + n0))[pv] = v;
    }
}

// ---------------------------------------------------------------------------
extern "C" void kernel_launch(void* const* d_in, const int* in_sizes, int n_in,
                              void* d_out, int out_size, void* d_ws, size_t ws_size,
                              hipStream_t stream) {
    (void)in_sizes; (void)n_in; (void)out_size; (void)ws_size;
    const float* x1f = (const float*)d_in[0];
    const float* x2f = (const float*)d_in[1];
    const float* x1c = (const float*)d_in[2];
    const float* x2c = (const float*)d_in[3];
    float* out = (float*)d_out;

    char* ws = (char*)d_ws;
    const size_t VOX_BYTES = (size_t)BATCH * R3 * CH * sizeof(float);  // 64 MB
    const size_t CNT_BYTES = (size_t)BATCH * R3 * sizeof(float);       // 1 MB
    float* vox     = (float*)ws;
    float* cnt     = (float*)(ws + VOX_BYTES);
    float* stats   = (float*)(ws + VOX_BYTES + CNT_BYTES);             // 16*4 f32
    float* partial = (float*)(ws + VOX_BYTES + CNT_BYTES + 256);       // 256*4 f32

    k_partial_sum<<<256, 256, 0, stream>>>(x1c, x2c, partial);
    k_mean_final<<<1, 32, 0, stream>>>(partial, stats);
    k_partial_max<<<256, 256, 0, stream>>>(x1c, x2c, stats, partial);
    k_max_final<<<1, 32, 0, stream>>>(partial, stats);

    const int nzero = (BATCH * R3 * CH / 4 + 255) / 256;
    k_zero<<<nzero, 256, 0, stream>>>((float4*)vox, cnt);

    dim3 gs(NPTS / 64, BATCH);
    k_scatter<<<gs, 256, 0, stream>>>(x2f, x2c, stats, vox, cnt);
    k_gather<<<gs, 256, 0, stream>>>(x1f, x1c, stats, vox, cnt, out);
}